// RepSurfUmbrella_7138235646417
// MI455X (gfx1250) — compile-verified
//
#include <hip/hip_runtime.h>
#include <hip/hip_bf16.h>
#include <math.h>

typedef __attribute__((ext_vector_type(2))) float v2f;
typedef __attribute__((ext_vector_type(8))) float v8f;
typedef __attribute__((ext_vector_type(4))) unsigned int u32x4;
typedef __attribute__((ext_vector_type(4))) int i32x4;
typedef __attribute__((ext_vector_type(8))) int i32x8;

#define NPTS   4096
#define BATCH  8
#define KNB    9
#define CHUNK  2048
#define WAVES  8
#define TPB1   256
#define PI_F   3.14159265358979323846f

__device__ __forceinline__ float nan0(float v) { return (v != v) ? 0.0f : v; }

// ---------------------------------------------------------------------------
// TDM: async-DMA a (3 x rowLen) f32 tile (row stride srcStride elements) from
// global into contiguous LDS at lds_off. Builds the D# descriptor per
// CDNA5 ISA: group0 = {count, lds_addr, global_addr,
// type=2}; group1 = {data_size=4B, tensor_dim0/1, tile_dim0/1, dim0 stride}.
// Tracked by TENSORcnt. EXEC is ignored by TENSOR_LOAD_TO_LDS, so the caller
// must gate this with a wave-uniform branch (one wave issues).
// ---------------------------------------------------------------------------
__device__ __forceinline__ void tdm_load_rows3(const float* gsrc, unsigned int lds_off,
                                               int rowLen, int srcStride) {
  unsigned long long ga = (unsigned long long)(uintptr_t)gsrc;
  u32x4 g0;
  g0[0] = 1u;                                        // count=1 (valid), user mode
  g0[1] = lds_off;                                   // lds_addr (bytes)
  g0[2] = (unsigned int)(ga & 0xffffffffu);          // global_addr[31:0]
  g0[3] = (unsigned int)((ga >> 32) & 0x01ffffffu)   // global_addr[56:32]
          | (2u << 30);                              // type=2 ("image")
  i32x8 g1;
  g1[0] = 0x00020000;                                // data_size=2 (4B), mask=0
  g1[1] = (int)(((unsigned int)srcStride & 0xffffu) << 16);      // tensor_dim0[15:0]
  g1[2] = (int)((((unsigned int)srcStride >> 16) & 0xffffu)      // tensor_dim0[31:16]
          | (3u << 16));                                         // tensor_dim1=3
  g1[3] = (int)((unsigned int)rowLen << 16);         // tile_dim0 (tensor_dim1 hi=0)
  g1[4] = 3;                                         // tile_dim1=3, tile_dim2=0
  g1[5] = srcStride;                                 // tensor_dim0_stride[31:0]
  g1[6] = 0;                                         // stride0 hi, stride1 lo
  g1[7] = 0;
  i32x4 z4 = {0, 0, 0, 0};
#if __clang_major__ >= 23
  i32x8 z8 = {0, 0, 0, 0, 0, 0, 0, 0};
  __builtin_amdgcn_tensor_load_to_lds(g0, g1, z4, z4, z8, 0);
#else
  __builtin_amdgcn_tensor_load_to_lds(g0, g1, z4, z4, 0);
#endif
}

// ---------------------------------------------------------------------------
// Kernel 1: kNN (top-10) via V_WMMA_F32_16X16X4_F32 distance tiles.
// Each wave owns 16 query columns (B fragment in registers). Candidates are
// TDM-staged into SoA LDS in 2048-point chunks; each 16-candidate tile is one
// WMMA (loop unrolled x2 -> two WMMAs per iteration). Per-lane sorted top-10
// in registers; two half-wave lists per query merged via LDS. Writes 9
// neighbor indices per point (self excluded).
// ---------------------------------------------------------------------------
__global__ __launch_bounds__(TPB1) void knn_wmma_kernel(
    const float* __restrict__ x, int* __restrict__ nbr) {
  __shared__ float sxyz[3 * CHUNK];       // 24 KB: sx | sy | sz (TDM target)
  __shared__ float ssq[CHUNK];            //  8 KB: |p|^2
  __shared__ float md[WAVES][16][20];     // 10 KB merge dists
  __shared__ int   mi[WAVES][16][20];     // 10 KB merge indices

  float* sx = sxyz;
  float* sy = sxyz + CHUNK;
  float* sz = sxyz + 2 * CHUNK;

  const int tid  = threadIdx.x;
  const int wave = tid >> 5;
  const int lane = tid & 31;
  const int col  = lane & 15;   // query column (C-layout N = lane%16)
  const int half = lane >> 4;   // 0: rows 0..7 / K0,K1   1: rows 8..15 / K2,K3

  const int blocksPerBatch = NPTS / (WAVES * 16);   // 32
  const int b     = blockIdx.x / blocksPerBatch;
  const int qbase = (blockIdx.x % blocksPerBatch) * (WAVES * 16);
  const int q     = qbase + wave * 16 + col;

  const float* X = x + (size_t)b * 3 * NPTS;

  // B fragment: 4x16 queries. K0=x,K1=y,K2=z,K3=0.
  v2f bfr;
  bfr.x = half ? X[2 * NPTS + q] : X[0 * NPTS + q];
  bfr.y = half ? 0.0f            : X[1 * NPTS + q];

  float topd[10];
  int   topi[10];
#pragma unroll
  for (int t = 0; t < 10; ++t) { topd[t] = 3.0e38f; topi[t] = 0x7fffffff; }

  const unsigned int lds_base = (unsigned int)(uintptr_t)&sxyz[0];

  for (int base = 0; base < NPTS; base += CHUNK) {
    if (tid < 32) {  // wave-uniform: wave 0 issues the DMA and drains TENSORcnt
      tdm_load_rows3(X + base, lds_base, CHUNK, NPTS);
      __builtin_amdgcn_s_wait_tensorcnt(0);
    }
    // overlap: all threads compute |p|^2 from global while the TDM runs
    for (int t = tid; t < CHUNK; t += TPB1) {
      float px = X[0 * NPTS + base + t];
      float py = X[1 * NPTS + base + t];
      float pz = X[2 * NPTS + base + t];
      ssq[t] = px * px + py * py + pz * pz;
    }
    __syncthreads();

    for (int jt = 0; jt < CHUNK; jt += 32) {
#pragma unroll
      for (int u = 0; u < 2; ++u) {
        const int jb = jt + u * 16;
        const int idx = jb + col;
        // A fragment: 16x4 candidates, row M = lane&15.
        // Low half needs (x,y) = (K0,K1); high half needs (z,0) = (K2,K3).
        const float* pa = half ? (sz + idx) : (sx + idx);
        v2f afr;
        afr.x = *pa;
        afr.y = half ? 0.0f : sy[idx];

        v8f c = {0.f, 0.f, 0.f, 0.f, 0.f, 0.f, 0.f, 0.f};
        c = __builtin_amdgcn_wmma_f32_16x16x4_f32(
            /*neg_a=*/false, afr, /*neg_b=*/false, bfr,
            /*c_mod=*/(short)0, c, /*reuse_a=*/false, /*reuse_b=*/false);

        // ranking key: |p_j|^2 - 2 p_i.p_j  (query norm is row-constant)
        const float* sqp = &ssq[jb + half * 8];
        float4 sq0 = *(const float4*)(sqp);
        float4 sq1 = *(const float4*)(sqp + 4);
        float sq8[8] = {sq0.x, sq0.y, sq0.z, sq0.w, sq1.x, sq1.y, sq1.z, sq1.w};

#pragma unroll
        for (int r = 0; r < 8; ++r) {
          float d  = sq8[r] - 2.0f * c[r];
          int   id = base + jb + half * 8 + r;
          if (d < topd[9]) {           // strict < keeps lower index on ties
            topd[9] = d; topi[9] = id;
#pragma unroll
            for (int s = 9; s > 0; --s) {  // single bubble pass re-sorts
              bool sw  = topd[s] < topd[s - 1];
              float dA = sw ? topd[s - 1] : topd[s];
              float dB = sw ? topd[s]     : topd[s - 1];
              int   iA = sw ? topi[s - 1] : topi[s];
              int   iB = sw ? topi[s]     : topi[s - 1];
              topd[s] = dA; topd[s - 1] = dB;
              topi[s] = iA; topi[s - 1] = iB;
            }
          }
        }
      }
    }
    __syncthreads();
  }

  // merge the two half-wave lists per query column
#pragma unroll
  for (int t = 0; t < 10; ++t) {
    md[wave][col][half * 10 + t] = topd[t];
    mi[wave][col][half * 10 + t] = topi[t];
  }
  __syncthreads();

  if (half == 0) {
    int a = 0, bb = 10;
    int* outp = nbr + ((size_t)b * NPTS + q) * KNB;
#pragma unroll
    for (int t = 0; t < 10; ++t) {
      float da = (a  < 10) ? md[wave][col][a]  : 3.4e38f;
      float db = (bb < 20) ? md[wave][col][bb] : 3.4e38f;
      int   ia = (a  < 10) ? mi[wave][col][a]  : 0x7fffffff;
      int   ib = (bb < 20) ? mi[wave][col][bb] : 0x7fffffff;
      bool takeA = (da < db) || (da == db && ia <= ib);
      int sel = takeA ? ia : ib;
      if (takeA) ++a; else ++bb;
      if (t > 0) outp[t - 1] = sel;   // entry 0 is self (global minimum)
    }
  }
}

// ---------------------------------------------------------------------------
// Kernel 2: umbrella features + folded-BN 9x9x3 MLP, sum over 9 triangles.
// One thread per (b, n). Deliberately scalar: a 9x9 matvec (K=9) would waste
// >70% of a 16x16x4 WMMA tile and need an LDS transpose per layer; total MLP
// work is ~145 MFLOP, so VALU FMA is the right path.
// ---------------------------------------------------------------------------
__global__ __launch_bounds__(256) void umbrella_mlp_kernel(
    const float* __restrict__ x, const int* __restrict__ nbr,
    const float* __restrict__ w1,
    const float* __restrict__ g1, const float* __restrict__ b1,
    const float* __restrict__ m1, const float* __restrict__ v1,
    const float* __restrict__ w2, const float* __restrict__ bias2,
    const float* __restrict__ g2, const float* __restrict__ b2,
    const float* __restrict__ m2, const float* __restrict__ v2,
    const float* __restrict__ w3, const float* __restrict__ bias3,
    float* __restrict__ out) {
  __shared__ float A1[81], A2[81], A3[81], C1[9], C2[9], C3[9];
  const int t = threadIdx.x;
  if (t < 81) {
    int o = t / 9;
    A1[t] = g1[o] * rsqrtf(v1[o] + 1e-5f) * w1[t];
    A2[t] = g2[o] * rsqrtf(v2[o] + 1e-5f) * w2[t];
    A3[t] = w3[t];
  }
  if (t < 9) {
    float s1 = g1[t] * rsqrtf(v1[t] + 1e-5f);
    float s2 = g2[t] * rsqrtf(v2[t] + 1e-5f);
    C1[t] = b1[t] - m1[t] * s1;
    C2[t] = b2[t] - m2[t] * s2 + s2 * bias2[t];
    C3[t] = bias3[t];
  }
  __syncthreads();

  const int gidx = blockIdx.x * blockDim.x + threadIdx.x;   // b*N + n
  const int b = gidx / NPTS, n = gidx % NPTS;
  const float* X = x + (size_t)b * 3 * NPTS;
  const float pix = X[n], piy = X[NPTS + n], piz = X[2 * NPTS + n];

  float gx[9], gy[9], gz[9], ph[9];
  const int* nb = nbr + (size_t)gidx * KNB;
#pragma unroll
  for (int k = 0; k < 9; ++k) {
    int j = nb[k];
    gx[k] = X[j] - pix;
    gy[k] = X[NPTS + j] - piy;
    gz[k] = X[2 * NPTS + j] - piz;
    ph[k] = atan2f(gy[k], gx[k]);   // monotonic in reference's phi
  }

  // stable bubble sort by azimuth, constant indices -> registers
#pragma unroll
  for (int pass = 0; pass < 9; ++pass) {
#pragma unroll
    for (int i = 0; i < 8; ++i) {
      if (ph[i] > ph[i + 1]) {
        float tp = ph[i]; ph[i] = ph[i + 1]; ph[i + 1] = tp;
        float tx = gx[i]; gx[i] = gx[i + 1]; gx[i + 1] = tx;
        float ty = gy[i]; gy[i] = gy[i + 1]; gy[i + 1] = ty;
        float tz = gz[i]; gz[i] = gz[i + 1]; gz[i + 1] = tz;
      }
    }
  }

  float acc[9];
#pragma unroll
  for (int o = 0; o < 9; ++o) acc[o] = 0.0f;

  float pos = 1.0f;
#pragma unroll
  for (int k = 0; k < 9; ++k) {
    const int k2 = (k + 1) % 9;
    float ax = gx[k],  ay = gy[k],  az = gz[k];
    float bx = gx[k2], by = gy[k2], bz = gz[k2];
    float cnx = ay * bz - az * by;
    float cny = az * bx - ax * bz;
    float cnz = ax * by - ay * bx;
    float nrm = sqrtf(cnx * cnx + cny * cny + cnz * cnz);
    float inv = 1.0f / fmaxf(nrm, 1e-10f);
    float ux = cnx * inv, uy = cny * inv, uz = cnz * inv;
    if (k == 0) pos = (ux > 0.0f) ? 1.0f : -1.0f;
    float nx = nan0(ux * pos), ny = nan0(uy * pos), nz = nan0(uz * pos);

    float cx = nan0((ax + bx) * (1.0f / 3.0f));
    float cy = nan0((ay + by) * (1.0f / 3.0f));
    float cz = nan0((az + bz) * (1.0f / 3.0f));
    float rho = sqrtf(cx * cx + cy * cy + cz * cz);
    float rs  = fmaxf(rho, 1e-8f);
    float th  = acosf(fminf(fmaxf(cz / rs, -1.0f), 1.0f)) * (1.0f / PI_F);
    float phc = atan2f(cy, cx) * (1.0f / (2.0f * PI_F)) + 0.5f;

    float f[9] = {cx, cy, cz, rho, th, phc, nx, ny, nz};
    float h1[9], h2[9];
#pragma unroll
    for (int o = 0; o < 9; ++o) {
      float s = C1[o];
#pragma unroll
      for (int cc = 0; cc < 9; ++cc) s += A1[o * 9 + cc] * f[cc];
      h1[o] = fmaxf(s, 0.0f);
    }
#pragma unroll
    for (int o = 0; o < 9; ++o) {
      float s = C2[o];
#pragma unroll
      for (int cc = 0; cc < 9; ++cc) s += A2[o * 9 + cc] * h1[cc];
      h2[o] = fmaxf(s, 0.0f);
    }
#pragma unroll
    for (int o = 0; o < 9; ++o) {
      float s = C3[o];
#pragma unroll
      for (int cc = 0; cc < 9; ++cc) s += A3[o * 9 + cc] * h2[cc];
      acc[o] += s;
    }
  }

  float* op = out + (size_t)b * 9 * NPTS + n;
#pragma unroll
  for (int o = 0; o < 9; ++o) op[o * NPTS] = acc[o];
}

// ---------------------------------------------------------------------------
extern "C" void kernel_launch(void* const* d_in, const int* in_sizes, int n_in,
                              void* d_out, int out_size, void* d_ws, size_t ws_size,
                              hipStream_t stream) {
  const float* x     = (const float*)d_in[0];
  const float* w1    = (const float*)d_in[1];
  const float* g1    = (const float*)d_in[2];
  const float* b1    = (const float*)d_in[3];
  const float* m1    = (const float*)d_in[4];
  const float* v1    = (const float*)d_in[5];
  const float* w2    = (const float*)d_in[6];
  const float* bias2 = (const float*)d_in[7];
  const float* g2    = (const float*)d_in[8];
  const float* b2    = (const float*)d_in[9];
  const float* m2    = (const float*)d_in[10];
  const float* v2    = (const float*)d_in[11];
  const float* w3    = (const float*)d_in[12];
  const float* bias3 = (const float*)d_in[13];
  float* out = (float*)d_out;
  int* nbr = (int*)d_ws;   // B*N*9 ints = 1.18 MB

  knn_wmma_kernel<<<BATCH * (NPTS / (WAVES * 16)), TPB1, 0, stream>>>(x, nbr);
  umbrella_mlp_kernel<<<(BATCH * NPTS) / 256, 256, 0, stream>>>(
      x, nbr, w1, g1, b1, m1, v1, w2, bias2, g2, b2, m2, v2, w3, bias3, out);
}